// CannyDetector_69114613727515
// MI455X (gfx1250) — compile-verified
//
#include <hip/hip_runtime.h>
#include <cstdint>

// CDNA5 (gfx1250) fused Canny edge detector.
// One kernel, one pass over HBM: ~64 MB traffic => ~2.8us roofline at 23.3 TB/s.
// - async global->LDS staging (global_load_async_to_lds_b32 / s_wait_asynccnt)
// - separable Gaussian blur done in full f32 via V_WMMA_F32_16X16X4_F32

typedef __attribute__((ext_vector_type(2))) float v2f;
typedef __attribute__((ext_vector_type(8))) float v8f;

#define IMG_N 2048
#define TILE  32
// LDS array strides (padded to odd to avoid bank conflicts)
#define IN_S 53   // staged input: 52x52 valid (rows 52..63 unused scratch), 64 x IN_S
#define HB_S 49   // h-blur:      64 rows x 48 cols valid (rows 52..63 garbage, never read)
#define BL_S 49   // blurred:     48 x 48
#define GM_S 37   // grad mag:    36 x 36  (tile +-2)
#define GS_S 35   // gx/gy sums:  34 x 34  (tile +-1)
#define TH_S 35   // thin:        34 x 34  (tile +-1)

// Unnormalized gaussian taps exp(-0.5*n^2), n=-2..2 (scipy.signal.gaussian, std=1)
__device__ __forceinline__ float gtap(int d) {
  if (d == 2)           return 1.0f;
  if (d == 1 || d == 3) return 0.60653065971263342f;
  if (d == 0 || d == 4) return 0.13533528323661270f;
  return 0.0f;
}

__global__ __launch_bounds__(256)
void canny_fused(const float* __restrict__ img,
                 const int*   __restrict__ t1p,
                 const int*   __restrict__ t2p,
                 float*       __restrict__ out)
{
  __shared__ float in_s[64 * IN_S];
  __shared__ float hb_s[64 * HB_S];
  __shared__ float bl_s[48 * BL_S];
  __shared__ float gm_s[36 * GM_S];
  __shared__ float gxs_s[34 * GS_S];
  __shared__ float gys_s[34 * GS_S];
  __shared__ float th_s[34 * TH_S];

  const int tid  = threadIdx.x;
  const int lane = tid & 31;
  const int wave = tid >> 5;
  const int frow = lane & 15;   // M (A-frag) / N (B,D-frags)
  const int hif  = lane >> 4;   // lane-half select
  const int fkb  = hif << 1;    // K pair base within a K=4 step

  const int gx0 = blockIdx.x * TILE;
  const int gy0 = blockIdx.y * TILE;

  const float ta = (float)t1p[0];
  const float tb = (float)t2p[0];
  const float lowt  = fminf(ta, tb);
  const float hight = fmaxf(ta, tb);

  // Constant band-matrix fragments. H-pass B[k][n]=g[k-n] and V-pass A[m][k]=g[k-m]
  // have identical per-lane values under the (documented-symmetric) layouts.
  v2f gfrag[5];
#pragma unroll
  for (int j = 0; j < 5; ++j) {
    const int k0 = 4 * j + fkb;
    gfrag[j].x = gtap(k0 - frow);
    gfrag[j].y = gtap(k0 + 1 - frow);
  }

  // zero cross-channel accumulators
  for (int i = tid; i < 36 * GM_S; i += 256) gm_s[i] = 0.f;
  for (int i = tid; i < 34 * GS_S; i += 256) { gxs_s[i] = 0.f; gys_s[i] = 0.f; }
  __syncthreads();

  const bool interior = (gy0 >= 10) && (gx0 >= 10) &&
                        (gy0 + 41 < IMG_N) && (gx0 + 41 < IMG_N);
  // Low 32 bits of a flat LDS pointer are the LDS byte offset (CDNA5 aperture rule).
  const uint32_t in_lds_base = (uint32_t)(size_t)(&in_s[0]);

  for (int ch = 0; ch < 3; ++ch) {
    const float* src = img + (size_t)ch * IMG_N * IMG_N;

    // ---------------- stage 52x52 input region (zero-padded at image edges) ----
    if (interior) {
      for (int i = tid; i < 52 * 52; i += 256) {
        const int r = i / 52, c = i - r * 52;
        const uint32_t ldso = in_lds_base + (uint32_t)(r * IN_S + c) * 4u;
        const uint32_t go = (uint32_t)(((gy0 - 10 + r) * IMG_N + (gx0 - 10 + c)) * 4);
        asm volatile("global_load_async_to_lds_b32 %0, %1, %2 offset:0"
                     :: "v"(ldso), "v"(go), "s"(src) : "memory");
      }
      asm volatile("s_wait_asynccnt 0" ::: "memory");
    } else {
      for (int i = tid; i < 52 * 52; i += 256) {
        const int r = i / 52, c = i - r * 52;
        const int Y = gy0 - 10 + r, X = gx0 - 10 + c;
        float v = 0.f;
        if ((unsigned)Y < (unsigned)IMG_N && (unsigned)X < (unsigned)IMG_N)
          v = src[Y * IMG_N + X];
        in_s[r * IN_S + c] = v;
      }
    }
    __syncthreads();

    // ---------------- horizontal blur: hb(16x16) = in(16x20) * B(20x16), f32 WMMA
    // 12 tiles (4 row-chunks x 3 col-chunks); rows 52..63 are garbage, never read.
    for (int t = wave; t < 12; t += 8) {
      const int R = t / 3, C = t - 3 * R;
      v8f acc = {0.f, 0.f, 0.f, 0.f, 0.f, 0.f, 0.f, 0.f};
      const int arow = R * 16 + frow;
#pragma unroll
      for (int j = 0; j < 5; ++j) {
        const int kk = C * 16 + 4 * j + fkb;
        v2f a;
        a.x = in_s[arow * IN_S + kk];
        a.y = in_s[arow * IN_S + kk + 1];
        acc = __builtin_amdgcn_wmma_f32_16x16x4_f32(false, a, false, gfrag[j],
                                                    (short)0, acc, false, false);
      }
#pragma unroll
      for (int v = 0; v < 8; ++v)
        hb_s[(R * 16 + v + 8 * hif) * HB_S + C * 16 + frow] = acc[v];
    }
    __syncthreads();

    // ---------------- vertical blur: bl(16x16) = G(16x20) * hb(20x16), f32 WMMA
    for (int t = wave; t < 9; t += 8) {
      const int R = t / 3, C = t - 3 * R;
      v8f acc = {0.f, 0.f, 0.f, 0.f, 0.f, 0.f, 0.f, 0.f};
#pragma unroll
      for (int j = 0; j < 5; ++j) {
        const int kr = R * 16 + 4 * j + fkb;
        v2f b;
        b.x = hb_s[kr * HB_S + C * 16 + frow];
        b.y = hb_s[(kr + 1) * HB_S + C * 16 + frow];
        acc = __builtin_amdgcn_wmma_f32_16x16x4_f32(false, gfrag[j], false, b,
                                                    (short)0, acc, false, false);
      }
#pragma unroll
      for (int v = 0; v < 8; ++v)
        bl_s[(R * 16 + v + 8 * hif) * BL_S + C * 16 + frow] = acc[v];
    }
    __syncthreads();

    // ---------------- Sobel + accumulate |grad| (sum over channels), gx/gy sums
    for (int i = tid; i < 36 * 36; i += 256) {
      const int ii = i / 36, jj = i - ii * 36;      // gmag coords (tile offset -2)
      const int r = ii + 6, c = jj + 6;             // center in bl_s
      const float tl = bl_s[(r - 1) * BL_S + c - 1];
      const float tc = bl_s[(r - 1) * BL_S + c];
      const float tr = bl_s[(r - 1) * BL_S + c + 1];
      const float ml = bl_s[r * BL_S + c - 1];
      const float mr = bl_s[r * BL_S + c + 1];
      const float bL = bl_s[(r + 1) * BL_S + c - 1];
      const float bc = bl_s[(r + 1) * BL_S + c];
      const float br = bl_s[(r + 1) * BL_S + c + 1];
      float gx = (tl - tr) + 2.f * (ml - mr) + (bL - br);
      float gy = (tl - bL) + 2.f * (tc - bc) + (tr - br);
      float mag = sqrtf(gx * gx + gy * gy);
      const int Y = gy0 + ii - 2, X = gx0 + jj - 2;
      if (!((unsigned)Y < (unsigned)IMG_N && (unsigned)X < (unsigned)IMG_N)) {
        gx = 0.f; gy = 0.f; mag = 0.f;   // match reference zero-padding of grad_mag
      }
      gm_s[ii * GM_S + jj] += mag;
      if (ii >= 1 && ii <= 34 && jj >= 1 && jj <= 34) {
        gxs_s[(ii - 1) * GS_S + (jj - 1)] += gx;
        gys_s[(ii - 1) * GS_S + (jj - 1)] += gy;
      }
    }
    __syncthreads();
  }

  // ---------------- orientation quantization + NMS -> thin (tile +-1 region)
  for (int i = tid; i < 34 * 34; i += 256) {
    const int u = i / 34, v = i - u * 34;
    const float sx = gxs_s[u * GS_S + v];
    const float sy = gys_s[u * GS_S + v];
    const float ang = atan2f(sy, sx) * 57.295779513082323f;
    const int q = ((int)roundf((ang + 180.f) * (1.f / 45.f))) & 7;
    // dy={0,1,1,1,0,-1,-1,-1}[q], dx={1,1,0,-1,-1,-1,0,1}[q], packed 2-bit (+1)
    const int dy = ((0x1A9  >> (2 * q)) & 3) - 1;
    const int dx = ((0x901A >> (2 * q)) & 3) - 1;
    const float m0 = gm_s[(u + 1) * GM_S + (v + 1)];
    const float mp = gm_s[(u + 1 + dy) * GM_S + (v + 1 + dx)];
    const float mn = gm_s[(u + 1 - dy) * GM_S + (v + 1 - dx)];
    float t = (fminf(m0 - mp, m0 - mn) > 0.f) ? m0 : 0.f;
    const int Y = gy0 + u - 1, X = gx0 + v - 1;
    if (!((unsigned)Y < (unsigned)IMG_N && (unsigned)X < (unsigned)IMG_N)) t = 0.f;
    th_s[u * TH_S + v] = t;
  }
  __syncthreads();

  // ---------------- hysteresis (single pass) + border zero + binarize
  for (int i = tid; i < TILE * TILE; i += 256) {
    const int oy = i >> 5, ox = i & 31;
    const int u = oy + 1, v = ox + 1;
    const float t = th_s[u * TH_S + v];
    float conn = 0.f;
#pragma unroll
    for (int dy = -1; dy <= 1; ++dy)
#pragma unroll
      for (int dx = -1; dx <= 1; ++dx) {
        if (dy == 0 && dx == 0) continue;
        conn += (th_s[(u + dy) * TH_S + (v + dx)] > hight) ? 1.f : 0.f;
      }
    const int Y = gy0 + oy, X = gx0 + ox;
    const bool edge = (Y == 0) || (Y == IMG_N - 1) || (X == 0) || (X == IMG_N - 1);
    const bool strong = t > hight;
    const bool mid    = (t >= lowt) && (t <= hight) && (conn > 0.f);
    out[Y * IMG_N + X] = (!edge && (strong || mid)) ? 1.f : 0.f;
  }
}

extern "C" void kernel_launch(void* const* d_in, const int* in_sizes, int n_in,
                              void* d_out, int out_size, void* d_ws, size_t ws_size,
                              hipStream_t stream) {
  (void)in_sizes; (void)n_in; (void)out_size; (void)d_ws; (void)ws_size;
  const float* img = (const float*)d_in[0];
  const int*   t1  = (const int*)d_in[1];
  const int*   t2  = (const int*)d_in[2];
  float* out = (float*)d_out;
  dim3 grid(IMG_N / TILE, IMG_N / TILE);  // 64 x 64 tiles
  dim3 block(256);
  hipLaunchKernelGGL(canny_fused, grid, block, 0, stream, img, t1, t2, out);
}